// ResidualBlock_cg_42090679501110
// MI455X (gfx1250) — compile-verified
//
#include <hip/hip_runtime.h>
#include <hip/hip_bf16.h>
#include <math.h>

// Problem constants (from reference):
#define NPTS  4096          // N points
#define NBF   9             // (L+1)^2 spherical basis count
#define CDIM  256           // channels
#define MIDD  64            // C/4
#define KNBR  16            // neighbors
#define MROWS (NPTS * NBF)  // 36864 rows of the projection GEMM

typedef __attribute__((ext_vector_type(2))) float v2f;
typedef __attribute__((ext_vector_type(8))) float v8f;

typedef __attribute__((address_space(3))) float lds_f32;
typedef const __attribute__((address_space(1))) float glb_f32;

__device__ __forceinline__ v8f wmma_f32_16x16x4(v2f a, v2f b, v8f c) {
  // D(16x16,f32) = A(16x4,f32) x B(4x16,f32) + C
  return __builtin_amdgcn_wmma_f32_16x16x4_f32(
      /*neg_a=*/false, a, /*neg_b=*/false, b,
      /*c_mod=*/(short)0, c, /*reuse_a=*/false, /*reuse_b=*/false);
}

// Async copy of 16 bytes global -> LDS (tracked by ASYNCcnt, no VGPR returns).
__device__ __forceinline__ void async_g2l_b128(lds_f32* lds_dst, glb_f32* gsrc) {
  asm volatile("global_load_async_to_lds_b128 %0, %1, off"
               :: "v"(lds_dst), "v"(gsrc) : "memory");
}
__device__ __forceinline__ void wait_asynccnt0() {
  asm volatile("s_wait_asynccnt 0" ::: "memory");
}

// -------------------------------------------------------------------------
// Kernel 1: h = reshape(s_feats, [36864,256]) @ W_proj (256x256), f32 WMMA.
// Block = 4 waves sharing one 32-row A block staged in LDS via
// GLOBAL_LOAD_ASYNC_TO_LDS_B128 (A global traffic cut 4x). Each wave owns a
// 32x64 output block (2 row-tiles x 4 col-tiles = 8 accumulators): per
// k-step 2 A fragments (LDS) + 4 B fragments (L2) feed 8 WMMAs.
// -------------------------------------------------------------------------
__global__ __launch_bounds__(128) void proj_gemm_kernel(
    const float* __restrict__ SF, const float* __restrict__ WP,
    float* __restrict__ H) {
  __shared__ float As[32 * 260];            // padded stride: conflict-free K-column reads

  const int tid  = threadIdx.x;
  const int lane = tid & 31;
  const int wave = tid >> 5;
  const int mt = blockIdx.x;                // 1152 row blocks (32 rows each)
  const int m0 = mt * 32;
  const int n0 = wave * 64;                 // 4 waves cover all 256 columns
  const int lh = lane & 15;
  const int kh = (lane >> 4) << 1;          // 0 or 2 (K sub-offset per ISA layout)

  // ---- stage A(32x256) into LDS asynchronously (16B per op) --------------
  {
    glb_f32* gbase = (glb_f32*)(SF + (size_t)m0 * CDIM);
    for (int v = tid; v < 32 * 64; v += 128) {          // 2048 float4 tiles
      const int r = v >> 6, c4 = (v & 63) << 2;
      async_g2l_b128((lds_f32*)&As[r * 260 + c4], gbase + (size_t)r * CDIM + c4);
    }
    wait_asynccnt0();
  }
  __syncthreads();

  v8f acc[8] = {};                          // [col 0..3] rows 0-15, [4+col] rows 16-31
  for (int k = 0; k < CDIM; k += 4) {
    v2f a0, a1;
    a0.x = As[lh * 260 + k + kh];        a0.y = As[lh * 260 + k + kh + 1];
    a1.x = As[(16 + lh) * 260 + k + kh]; a1.y = As[(16 + lh) * 260 + k + kh + 1];
    const float* __restrict__ b0p = WP + (size_t)(k + kh) * CDIM + n0 + lh;
    const float* __restrict__ b1p = WP + (size_t)(k + kh + 1) * CDIM + n0 + lh;
    #pragma unroll
    for (int c = 0; c < 4; ++c) {
      v2f b; b.x = b0p[c * 16]; b.y = b1p[c * 16];
      acc[c]     = wmma_f32_16x16x4(a0, b, acc[c]);
      acc[4 + c] = wmma_f32_16x16x4(a1, b, acc[4 + c]);
    }
  }
  // D layout: VGPR r -> row r (lanes 0-15) / row r+8 (lanes 16-31), col = lane%16
  const int ro = (lane >> 4) << 3;
  #pragma unroll
  for (int c = 0; c < 4; ++c) {
    float* __restrict__ h0 = H + (size_t)(m0 + ro) * CDIM + n0 + c * 16 + lh;
    float* __restrict__ h1 = h0 + (size_t)16 * CDIM;
    #pragma unroll
    for (int r = 0; r < 8; ++r) {
      h0[(size_t)r * CDIM] = acc[c][r];
      h1[(size_t)r * CDIM] = acc[4 + c][r];
    }
  }
}

// -------------------------------------------------------------------------
// Kernel 2 (fused, one block of 256 threads per point n):
//   sh/cg -> A(16x144); B rows gathered from h (L2-resident, prefetched) ->
//   x = A@B + s_feats (WMMA)   [x kept in LDS = shortcut]
//   per-degree norms -> y ; z = y @ W_ff1 (WMMA) ; gate GEMV ; gating ;
//   out = z_g @ W_ff2 + x (WMMA)
// -------------------------------------------------------------------------
__global__ __launch_bounds__(256) void fused_block_kernel(
    const float* __restrict__ SF, const float* __restrict__ PTS,
    const float* __restrict__ CG, const float* __restrict__ LNW,
    const float* __restrict__ LNB, const float* __restrict__ WF1,
    const float* __restrict__ WG, const float* __restrict__ WF2,
    const int* __restrict__ IDX, const float* __restrict__ H,
    float* __restrict__ OUT) {
  const int n    = blockIdx.x;
  const int tid  = threadIdx.x;
  const int lane = tid & 31;
  const int wave = tid >> 5;
  const int lh   = lane & 15;
  const int kh   = (lane >> 4) << 1;
  const int rb   = (lane >> 4) << 3;

  __shared__ float    sh_lds[KNBR * 12];     // 9 SH values per neighbor (padded)
  __shared__ int      idx_lds[KNBR];
  __shared__ unsigned rowbase[144];          // gathered h row offsets (floats)
  __shared__ float    Aw[16 * 148];          // A matrix (o x ki), padded stride
  __shared__ float    x_lds[16 * 256];       // x = CG output + residual (shortcut)
  __shared__ float    y_lds[16 * 260];       // normalized, padded stride
  __shared__ float    z_lds[16 * 68];        // ff1 output / gated, padded stride
  __shared__ float    gate_lds[192];
  __shared__ float    red0[256], red1[256], red2[256], red3[256];

  // ---- 1) neighbor indices + real SH(l<=2) -------------------------------
  if (tid < KNBR) {
    const int nb = IDX[n * KNBR + tid];
    idx_lds[tid] = nb;
    const float rx = PTS[nb * 3 + 0] - PTS[n * 3 + 0];
    const float ry = PTS[nb * 3 + 1] - PTS[n * 3 + 1];
    const float rz = PTS[nb * 3 + 2] - PTS[n * 3 + 2];
    const float r  = sqrtf(rx * rx + ry * ry + rz * rz);
    const float iv = 1.0f / (r + 1e-8f);
    const float x = rx * iv, y = ry * iv, z = rz * iv;
    const float c0 = 0.28209479177387814f, c1 = 0.4886025119029199f;
    const float c2 = 1.0925484305920792f, c2b = 0.31539156525252005f;
    const float c2c = 0.5462742152960396f;
    float* s = sh_lds + tid * 12;
    s[0] = c0;          s[1] = c1 * y;      s[2] = c1 * z;
    s[3] = c1 * x;      s[4] = c2 * x * y;  s[5] = c2 * y * z;
    s[6] = c2b * (3.0f * z * z - 1.0f);
    s[7] = c2 * x * z;  s[8] = c2c * (x * x - y * y);
  }
  __syncthreads();

  // ---- 2) gather row offsets + A[o][k*9+i] = (1/K) sum_j sh[k][j]*cg[i][j][o]
  if (tid < 144) {
    const int k = tid / 9, i = tid % 9;
    rowbase[tid] = (unsigned)((idx_lds[k] * NBF + i) * CDIM);
    // Warm L0/L2 for the indirect gather feeding the WMMA loop below
    // (lowered to global_prefetch_b8; whole cacheline pulled per row).
    __builtin_prefetch((const void*)(H + (size_t)rowbase[tid]), 0, 1);
  }
  for (int idx = tid; idx < 16 * 144; idx += 256) {
    const int o = idx / 144, ki = idx % 144;
    float acc = 0.0f;
    if (o < NBF) {
      const int k = ki / 9, i = ki % 9;
      const float* s = sh_lds + k * 12;
      #pragma unroll
      for (int j = 0; j < 9; ++j) acc += s[j] * CG[(i * 9 + j) * 9 + o];
    }
    Aw[o * 148 + ki] = acc * (1.0f / (float)KNBR);  // fold /K
  }
  __syncthreads();

  // ---- 3) x = A(16x144) @ Bgathered(144x256) + s_feats  (WMMA) -----------
  // Each wave: two channel tiles (c0, c0+128) sharing one A-fragment stream.
  {
    const int c0 = wave * 16;
    const int c1 = c0 + 128;
    v8f acc0 = {}, acc1 = {};
    for (int kk = 0; kk < 144; kk += 4) {
      v2f a, b0, b1;
      a.x = Aw[lh * 148 + kk + kh];
      a.y = Aw[lh * 148 + kk + kh + 1];
      const unsigned r0 = rowbase[kk + kh];
      const unsigned r1 = rowbase[kk + kh + 1];
      b0.x = H[(size_t)r0 + c0 + lh];
      b0.y = H[(size_t)r1 + c0 + lh];
      b1.x = H[(size_t)r0 + c1 + lh];
      b1.y = H[(size_t)r1 + c1 + lh];
      acc0 = wmma_f32_16x16x4(a, b0, acc0);
      acc1 = wmma_f32_16x16x4(a, b1, acc1);
    }
    #pragma unroll
    for (int r = 0; r < 8; ++r) {
      const int o = rb + r;
      float v0 = 0.0f, v1 = 0.0f;
      if (o < NBF) {
        const float* sf = SF + ((size_t)n * NBF + o) * CDIM;
        v0 = acc0[r] + sf[c0 + lh];
        v1 = acc1[r] + sf[c1 + lh];
      }
      x_lds[o * 256 + c0 + lh] = v0;  // rows 9..15 zero-padded for ff1
      x_lds[o * 256 + c1 + lh] = v1;
    }
  }
  __syncthreads();

  // ---- 4) per-degree norms -> y ------------------------------------------
  {
    const float v0 = x_lds[tid];   // row 0, channel tid
    float p2 = 0.0f, p3 = 0.0f;
    for (int rr = 1; rr < 4; ++rr) { const float v = x_lds[rr * 256 + tid]; p2 += v * v; }
    for (int rr = 4; rr < 9; ++rr) { const float v = x_lds[rr * 256 + tid]; p3 += v * v; }
    red0[tid] = v0; red1[tid] = v0 * v0; red2[tid] = p2; red3[tid] = p3;
  }
  __syncthreads();
  for (int s = 128; s > 0; s >>= 1) {
    if (tid < s) {
      red0[tid] += red0[tid + s]; red1[tid] += red1[tid + s];
      red2[tid] += red2[tid + s]; red3[tid] += red3[tid + s];
    }
    __syncthreads();
  }
  const float mu0 = red0[0] * (1.0f / 256.0f);
  const float sc0 = rsqrtf(red1[0] * (1.0f / 256.0f) - mu0 * mu0 + 1e-5f);
  const float sc1 = rsqrtf(red2[0] * (1.0f / 768.0f) + 1e-5f);
  const float sc2 = rsqrtf(red3[0] * (1.0f / 1280.0f) + 1e-5f);
  for (int idx = tid; idx < 16 * 256; idx += 256) {
    const int o = idx >> 8, c = idx & 255;
    const float xv = x_lds[o * 256 + c];
    float yv;
    if (o == 0)      yv = (xv - mu0) * sc0 * LNW[c] + LNB[c];
    else if (o < 4)  yv = xv * sc1 * LNW[256 + c];
    else if (o < 9)  yv = xv * sc2 * LNW[512 + c];
    else             yv = 0.0f;
    y_lds[o * 260 + c] = yv;
  }
  __syncthreads();

  // ---- 5) z = y(16x256) @ W_ff1(256x64)  (WMMA, waves 0..3) --------------
  if (wave < 4) {
    const int mc0 = wave * 16;
    v8f acc = {};
    for (int kk = 0; kk < CDIM; kk += 4) {
      v2f a, b;
      a.x = y_lds[lh * 260 + kk + kh];
      a.y = y_lds[lh * 260 + kk + kh + 1];
      b.x = WF1[(kk + kh) * MIDD + mc0 + lh];
      b.y = WF1[(kk + kh + 1) * MIDD + mc0 + lh];
      acc = wmma_f32_16x16x4(a, b, acc);
    }
    #pragma unroll
    for (int r = 0; r < 8; ++r) z_lds[(rb + r) * 68 + mc0 + lh] = acc[r];
  }
  __syncthreads();

  // ---- 6) gate vector: after = z[0,:] @ W_gate (64 -> 192) ---------------
  if (tid < 192) {
    float acc = 0.0f;
    for (int m = 0; m < MIDD; ++m) acc += z_lds[m] * WG[m * 192 + tid];
    gate_lds[tid] = acc;
  }
  __syncthreads();

  // ---- 7) gating: row0 = silu(after[:64]); rows 1..8 *= sigmoid(...) -----
  for (int idx = tid; idx < NBF * MIDD; idx += 256) {
    const int o = idx / MIDD, m = idx & (MIDD - 1);
    if (o == 0) {
      const float g = gate_lds[m];
      z_lds[m] = g / (1.0f + expf(-g));                       // silu
    } else {
      const float g = gate_lds[(o < 4 ? 64 : 128) + m];
      z_lds[o * 68 + m] *= 1.0f / (1.0f + expf(-g));          // sigmoid
    }
  }
  __syncthreads();

  // ---- 8) out = z_g(16x64) @ W_ff2(64x256) + shortcut  (WMMA) ------------
  // Each wave: two channel tiles (c0, c0+128) sharing one A-fragment stream.
  {
    const int c0 = wave * 16;
    const int c1 = c0 + 128;
    v8f acc0 = {}, acc1 = {};
    for (int kk = 0; kk < MIDD; kk += 4) {
      v2f a, b0, b1;
      a.x = z_lds[lh * 68 + kk + kh];
      a.y = z_lds[lh * 68 + kk + kh + 1];
      const float* __restrict__ w0 = WF2 + (size_t)(kk + kh) * CDIM + lh;
      const float* __restrict__ w1 = WF2 + (size_t)(kk + kh + 1) * CDIM + lh;
      b0.x = w0[c0]; b0.y = w1[c0];
      b1.x = w0[c1]; b1.y = w1[c1];
      acc0 = wmma_f32_16x16x4(a, b0, acc0);
      acc1 = wmma_f32_16x16x4(a, b1, acc1);
    }
    #pragma unroll
    for (int r = 0; r < 8; ++r) {
      const int o = rb + r;
      if (o < NBF) {
        float* __restrict__ op = OUT + ((size_t)n * NBF + o) * CDIM;
        op[c0 + lh] = acc0[r] + x_lds[o * 256 + c0 + lh];
        op[c1 + lh] = acc1[r] + x_lds[o * 256 + c1 + lh];
      }
    }
  }
}

// -------------------------------------------------------------------------
extern "C" void kernel_launch(void* const* d_in, const int* in_sizes, int n_in,
                              void* d_out, int out_size, void* d_ws, size_t ws_size,
                              hipStream_t stream) {
  const float* SF  = (const float*)d_in[0];   // s_feats (N,9,256)
  const float* PTS = (const float*)d_in[1];   // s_points (N,3)
  const float* CG  = (const float*)d_in[2];   // cg (9,9,9)
  const float* WP  = (const float*)d_in[3];   // W_proj (256,256)
  const float* LNW = (const float*)d_in[4];   // ln_w (3,256)
  const float* LNB = (const float*)d_in[5];   // ln_b (256)
  const float* WF1 = (const float*)d_in[6];   // W_ff1 (256,64)
  const float* WG  = (const float*)d_in[7];   // W_gate (64,192)
  const float* WF2 = (const float*)d_in[8];   // W_ff2 (64,256)
  const int*   IDX = (const int*)d_in[9];     // neighbor_indices (N,16)
  float* OUT = (float*)d_out;
  float* H   = (float*)d_ws;                  // h workspace: 36864*256 f32 (~37.7 MB)

  // one block (4 waves) per 32-row A block; waves split the 256 columns
  proj_gemm_kernel<<<dim3(MROWS / 32), dim3(128), 0, stream>>>(SF, WP, H);
  // one 256-thread block (8 waves) per point
  fused_block_kernel<<<dim3(NPTS), dim3(256), 0, stream>>>(
      SF, PTS, CG, LNW, LNB, WF1, WG, WF2, IDX, H, OUT);
}